// SelectiveSSM_51805895524532
// MI455X (gfx1250) — compile-verified
//
#include <hip/hip_runtime.h>
#include <hip/hip_bf16.h>

// Problem dims (from reference)
#define B_SZ   2
#define LSEQ   2048
#define DI     2048     // D_INNER
#define DS     16       // D_STATE
#define DTR    64       // DT_RANK
#define P_DIM  96       // DT_RANK + 2*D_STATE
#define BL     (B_SZ * LSEQ)   // 4096 rows for the projection GEMMs

// Chunked-scan config
#define NCHUNK 16
#define LC     (LSEQ / NCHUNK)  // 128

typedef float v2f __attribute__((ext_vector_type(2)));
typedef float v8f __attribute__((ext_vector_type(8)));
typedef int   v4i __attribute__((ext_vector_type(4)));

// CDNA5 async global->LDS path (guarded: also keeps the host pass compiling)
#if __has_builtin(__builtin_amdgcn_global_load_async_to_lds_b128) && \
    __has_builtin(__builtin_amdgcn_s_wait_asynccnt)
#define SSM_ASYNC_LDS 1
#endif

// ---------------------------------------------------------------------------
// Stage the B/C projection columns (cols [DTR, DTR+32) of xp) for one chunk
// into LDS: 128 rows x 32 floats = 16 KB, shared by all 8 waves of the block.
// Uses GLOBAL_LOAD_ASYNC_TO_LDS_B128 (ASYNCcnt) when available.
// ---------------------------------------------------------------------------
__device__ __forceinline__ void stage_bc_rows(const float* __restrict__ xp,
                                              int b, int t0,
                                              float (*sbc)[32]) {
    const int tid = threadIdx.x;
    #pragma unroll
    for (int u = 0; u < (LC * 8) / 256; ++u) {      // 4 x b128 per thread
        const int idx = u * 256 + tid;
        const int tr  = idx >> 3;                   // row within chunk
        const int seg = idx & 7;                    // 4-float segment
        float* src = (float*)(xp + (size_t)(b * LSEQ + t0 + tr) * P_DIM
                              + DTR + seg * 4);
#if defined(SSM_ASYNC_LDS)
        __builtin_amdgcn_global_load_async_to_lds_b128(
            (v4i*)src, (v4i*)&sbc[tr][seg * 4], 0, 0);
#else
        *(float4*)&sbc[tr][seg * 4] = *(const float4*)src;
#endif
    }
#if defined(SSM_ASYNC_LDS)
    __builtin_amdgcn_s_wait_asynccnt(0);
#endif
    __syncthreads();
}

// ---------------------------------------------------------------------------
// Kernel 1: xp = x (4096x2048) * W_x^T (2048x96)  via V_WMMA_F32_16X16X4_F32
// One wave per 16x16 output tile. 6 tiles along P (96), 256 tiles along M.
// ---------------------------------------------------------------------------
__global__ __launch_bounds__(256) void xproj_wmma(
    const float* __restrict__ x,      // (BL, DI) row-major
    const float* __restrict__ Wx,     // (P_DIM, DI) row-major
    float* __restrict__ xp)           // (BL, P_DIM)
{
    const int wave = threadIdx.x >> 5;
    const int lane = threadIdx.x & 31;
    const int tile = blockIdx.x * 8 + wave;     // 1536 tiles total
    const int tileM = tile / 6;
    const int tileP = tile % 6;

    const int kh = (lane >> 4) << 1;            // K sub-offset per half-wave
    const float* aptr = x  + (size_t)(tileM * 16 + (lane & 15)) * DI + kh;
    const float* bptr = Wx + (size_t)(tileP * 16 + (lane & 15)) * DI + kh;

    v8f c0 = {};
    v8f c1 = {};
    // K = 2048, 4 per WMMA -> 512 steps; dual accumulators for pipelining
    #pragma unroll 4
    for (int kb = 0; kb < DI / 8; ++kb) {
        v2f a0 = *(const v2f*)(aptr + kb * 8);
        v2f b0 = *(const v2f*)(bptr + kb * 8);
        v2f a1 = *(const v2f*)(aptr + kb * 8 + 4);
        v2f b1 = *(const v2f*)(bptr + kb * 8 + 4);
        c0 = __builtin_amdgcn_wmma_f32_16x16x4_f32(false, a0, false, b0,
                                                   (short)0, c0, false, false);
        c1 = __builtin_amdgcn_wmma_f32_16x16x4_f32(false, a1, false, b1,
                                                   (short)0, c1, false, false);
    }
    v8f c = c0 + c1;

    const int mo = tileM * 16 + (lane >> 4) * 8;   // row base for this lane
    const int po = tileP * 16 + (lane & 15);
    #pragma unroll
    for (int r = 0; r < 8; ++r)
        xp[(size_t)(mo + r) * P_DIM + po] = c[r];
}

// ---------------------------------------------------------------------------
// Kernel 2: delta = softplus( xp[:, :64] (4096x64) * W_dt^T (64x2048) + b_dt )
// ---------------------------------------------------------------------------
__global__ __launch_bounds__(256) void dtproj_wmma(
    const float* __restrict__ xp,     // (BL, P_DIM), use cols [0, DTR)
    const float* __restrict__ Wdt,    // (DI, DTR) row-major
    const float* __restrict__ bdt,    // (DI)
    float* __restrict__ delta)        // (BL, DI)
{
    const int wave = threadIdx.x >> 5;
    const int lane = threadIdx.x & 31;
    const int tile = blockIdx.x * 8 + wave;     // 256 * 128 = 32768 tiles
    const int tileM = tile >> 7;                // along BL
    const int tileD = tile & 127;               // along DI

    const int kh = (lane >> 4) << 1;
    const float* aptr = xp  + (size_t)(tileM * 16 + (lane & 15)) * P_DIM + kh;
    const float* bptr = Wdt + (size_t)(tileD * 16 + (lane & 15)) * DTR + kh;

    v8f c0 = {};
    v8f c1 = {};
    #pragma unroll
    for (int kb = 0; kb < DTR / 8; ++kb) {      // 8 double-steps (K=64)
        v2f a0 = *(const v2f*)(aptr + kb * 8);
        v2f b0 = *(const v2f*)(bptr + kb * 8);
        v2f a1 = *(const v2f*)(aptr + kb * 8 + 4);
        v2f b1 = *(const v2f*)(bptr + kb * 8 + 4);
        c0 = __builtin_amdgcn_wmma_f32_16x16x4_f32(false, a0, false, b0,
                                                   (short)0, c0, false, false);
        c1 = __builtin_amdgcn_wmma_f32_16x16x4_f32(false, a1, false, b1,
                                                   (short)0, c1, false, false);
    }
    v8f c = c0 + c1;

    const int mo = tileM * 16 + (lane >> 4) * 8;
    const int dcol = tileD * 16 + (lane & 15);
    const float bb = bdt[dcol];
    #pragma unroll
    for (int r = 0; r < 8; ++r) {
        float v = c[r] + bb;
        // softplus with overflow guard (matches jax.nn.softplus numerics)
        float sp = (v > 20.0f) ? v : log1pf(__expf(v));
        delta[(size_t)(mo + r) * DI + dcol] = sp;
    }
}

// ---------------------------------------------------------------------------
// Kernel 3 (phase 1): per-chunk local scan with h0=0; also per-chunk product
// of the decay factors.  h_t = a_t*h + dl*B_t*x ;  P = prod a_t
// One thread per (b, chunk, d); 16-state in registers; B rows from LDS.
// ---------------------------------------------------------------------------
__global__ __launch_bounds__(256) void scan_phase1(
    const float* __restrict__ x,       // (BL, DI)
    const float* __restrict__ A_log,   // (DI, DS)
    const float* __restrict__ xp,      // (BL, P_DIM): B at [DTR, DTR+DS)
    const float* __restrict__ delta,   // (BL, DI)
    float* __restrict__ Pbuf,          // (B*NCHUNK, DI, DS)
    float* __restrict__ Hloc)          // (B*NCHUNK, DI, DS)
{
    __shared__ float sbc[LC][32];              // B (0..15) and C (16..31) cols

    const int d  = blockIdx.y * 256 + threadIdx.x;
    const int bc = blockIdx.x;                 // b*NCHUNK + chunk
    const int b  = bc / NCHUNK;
    const int ch = bc % NCHUNK;
    const int t0 = ch * LC;

    stage_bc_rows(xp, b, t0, sbc);

    float nA[DS];
    #pragma unroll
    for (int n = 0; n < DS; ++n) nA[n] = -__expf(A_log[d * DS + n]);

    float h[DS], p[DS];
    #pragma unroll
    for (int n = 0; n < DS; ++n) { h[n] = 0.0f; p[n] = 1.0f; }

    for (int t = 0; t < LC; ++t) {
        const size_t row = (size_t)(b * LSEQ + t0 + t);
        // prefetch the strided row a few timesteps ahead (speculative)
        __builtin_prefetch(&delta[(row + 4) * DI + d], 0, 0);
        __builtin_prefetch(&x[(row + 4) * DI + d], 0, 0);
        const float dl = delta[row * DI + d];
        const float xv = x[row * DI + d];
        const float dx = dl * xv;
        #pragma unroll
        for (int n = 0; n < DS; ++n) {
            float a = __expf(dl * nA[n]);      // v_exp_f32 (TRANS, co-executes)
            h[n] = fmaf(a, h[n], dx * sbc[t][n]);
            p[n] *= a;
        }
    }
    const size_t base = ((size_t)bc * DI + d) * DS;
    #pragma unroll
    for (int n = 0; n < DS; ++n) { Pbuf[base + n] = p[n]; Hloc[base + n] = h[n]; }
}

// ---------------------------------------------------------------------------
// Kernel 4 (phase 2): sequential combine across the 16 chunks.
// One thread per (b, d, n) = 65536 threads. Emits the incoming state per chunk.
// ---------------------------------------------------------------------------
__global__ __launch_bounds__(256) void scan_phase2(
    const float* __restrict__ Pbuf,
    const float* __restrict__ Hloc,
    float* __restrict__ Hin)
{
    const int idx = blockIdx.x * 256 + threadIdx.x;   // b*DI*DS + d*DS + n
    const int b  = idx / (DI * DS);
    const int dn = idx % (DI * DS);
    float h = 0.0f;
    #pragma unroll
    for (int c = 0; c < NCHUNK; ++c) {
        const size_t off = (size_t)(b * NCHUNK + c) * (DI * DS) + dn;
        Hin[off] = h;
        h = fmaf(Pbuf[off], h, Hloc[off]);
    }
}

// ---------------------------------------------------------------------------
// Kernel 5 (phase 3): replay local scan with the correct incoming state and
// produce y = sum_n h[n]*C[n] + x*D.  B/C rows from LDS.
// ---------------------------------------------------------------------------
__global__ __launch_bounds__(256) void scan_phase3(
    const float* __restrict__ x,
    const float* __restrict__ A_log,
    const float* __restrict__ Dvec,    // (DI)
    const float* __restrict__ xp,      // B at [DTR,DTR+DS), C at [DTR+DS,DTR+2DS)
    const float* __restrict__ delta,
    const float* __restrict__ Hin,
    float* __restrict__ out)           // (BL, DI)
{
    __shared__ float sbc[LC][32];

    const int d  = blockIdx.y * 256 + threadIdx.x;
    const int bc = blockIdx.x;
    const int b  = bc / NCHUNK;
    const int ch = bc % NCHUNK;
    const int t0 = ch * LC;

    stage_bc_rows(xp, b, t0, sbc);

    float nA[DS];
    #pragma unroll
    for (int n = 0; n < DS; ++n) nA[n] = -__expf(A_log[d * DS + n]);

    float h[DS];
    const size_t base = ((size_t)bc * DI + d) * DS;
    #pragma unroll
    for (int n = 0; n < DS; ++n) h[n] = Hin[base + n];

    const float Dd = Dvec[d];
    for (int t = 0; t < LC; ++t) {
        const size_t row = (size_t)(b * LSEQ + t0 + t);
        __builtin_prefetch(&delta[(row + 4) * DI + d], 0, 0);
        __builtin_prefetch(&x[(row + 4) * DI + d], 0, 0);
        const float dl = delta[row * DI + d];
        const float xv = x[row * DI + d];
        const float dx = dl * xv;
        float y = 0.0f;
        #pragma unroll
        for (int n = 0; n < DS; ++n) {
            float a = __expf(dl * nA[n]);
            h[n] = fmaf(a, h[n], dx * sbc[t][n]);
            y = fmaf(h[n], sbc[t][16 + n], y);
        }
        out[row * DI + d] = fmaf(xv, Dd, y);
    }
}

// ---------------------------------------------------------------------------
// Launch
// ---------------------------------------------------------------------------
extern "C" void kernel_launch(void* const* d_in, const int* in_sizes, int n_in,
                              void* d_out, int out_size, void* d_ws, size_t ws_size,
                              hipStream_t stream) {
    const float* x     = (const float*)d_in[0];   // (B, L, DI)
    const float* A_log = (const float*)d_in[1];   // (DI, DS)
    const float* Dvec  = (const float*)d_in[2];   // (DI)
    const float* Wx    = (const float*)d_in[3];   // (P_DIM, DI)
    const float* Wdt   = (const float*)d_in[4];   // (DI, DTR)
    const float* bdt   = (const float*)d_in[5];   // (DI)
    float* out = (float*)d_out;

    // Workspace layout (floats)
    float* w      = (float*)d_ws;
    float* xp     = w;                                       // BL * P_DIM
    float* delta  = xp + (size_t)BL * P_DIM;                 // BL * DI
    float* Pbuf   = delta + (size_t)BL * DI;                 // B*NCHUNK*DI*DS
    float* Hloc   = Pbuf + (size_t)B_SZ * NCHUNK * DI * DS;
    float* Hin    = Hloc + (size_t)B_SZ * NCHUNK * DI * DS;

    // GEMM 1: 256 * 6 = 1536 wave-tiles, 8 waves per block
    xproj_wmma<<<dim3(1536 / 8), dim3(256), 0, stream>>>(x, Wx, xp);

    // GEMM 2: 256 * 128 = 32768 wave-tiles
    dtproj_wmma<<<dim3(32768 / 8), dim3(256), 0, stream>>>(xp, Wdt, bdt, delta);

    // Chunked scan
    scan_phase1<<<dim3(B_SZ * NCHUNK, DI / 256), dim3(256), 0, stream>>>(
        x, A_log, xp, delta, Pbuf, Hloc);
    scan_phase2<<<dim3((B_SZ * DI * DS) / 256), dim3(256), 0, stream>>>(
        Pbuf, Hloc, Hin);
    scan_phase3<<<dim3(B_SZ * NCHUNK, DI / 256), dim3(256), 0, stream>>>(
        x, A_log, Dvec, xp, delta, Hin, out);
}